// HCModule_90829968375849
// MI455X (gfx1250) — compile-verified
//
#include <hip/hip_runtime.h>
#include <hip/hip_bf16.h>
#include <math.h>

typedef __attribute__((ext_vector_type(2))) float v2f;
typedef __attribute__((ext_vector_type(8))) float v8f;

#define TOK  8192   // B*S
#define NSTR 4      // N streams
#define CH   1024   // C
#define DD   4096   // N*C
#define EPSR 1e-8f
#define KT   64     // K-chunk
#define STRD 68     // padded LDS row stride (68 % 64 == 4 banks -> conflict-free frag reads; 272B rows keep 16B chunk alignment)

// dynamic-LDS layout (floats)
#define A_BUF   (64 * STRD)            // 64 token rows x KT
#define B_BUF   (128 * STRD)           // 128 channel rows x KT
#define SM_A    0
#define SM_B    (2 * A_BUF)
#define SM_HP   (2 * A_BUF + 2 * B_BUF)
#define SM_HR   (SM_HP + 256)
#define SM_TOT  (SM_HR + 1024)
#define SMEM_BYTES (SM_TOT * 4)        // 109,568 B (< 320KB/WGP)

// CDNA5 async global->LDS copy, 16B per lane, GVS mode (SGPR64 base + VGPR byte offset).
// Tracked with ASYNCcnt; no VGPR data round-trip.
__device__ __forceinline__ void async_b128(unsigned lds_off, const float* base, unsigned byte_off) {
    asm volatile("global_load_async_to_lds_b128 %0, %1, %2"
                 :: "v"(lds_off), "v"(byte_off), "s"(base)
                 : "memory");
}
__device__ __forceinline__ void wait_async0() {
    asm volatile("s_wait_asynccnt 0x0" ::: "memory");
}

// ---------------------------------------------------------------------------
// Kernel 1: per-token fused RMSNorm + 24 tiny projections + tanh + h_in
// ---------------------------------------------------------------------------
__global__ __launch_bounds__(256) void hc_prep(
    const float* __restrict__ x, const float* __restrict__ norm_w,
    const float* __restrict__ w_pre, const float* __restrict__ w_post,
    const float* __restrict__ w_res,
    const float* __restrict__ a_pre, const float* __restrict__ b_pre,
    const float* __restrict__ a_post, const float* __restrict__ b_post,
    const float* __restrict__ a_res, const float* __restrict__ b_res,
    float* __restrict__ h_in, float* __restrict__ h_post_o,
    float* __restrict__ h_res_o)
{
    __shared__ float s_x[DD];
    __shared__ float s_red[25][8];
    __shared__ float s_scal[5];

    const int tid = threadIdx.x;
    const int t   = blockIdx.x;
    const size_t base = (size_t)t * DD;

    float acc[25];
#pragma unroll
    for (int i = 0; i < 25; ++i) acc[i] = 0.0f;

#pragma unroll
    for (int it = 0; it < 4; ++it) {
        const int idx = (tid + it * 256) * 4;
        const float4 xv = *(const float4*)(x + base + idx);
        const float4 nw = *(const float4*)(norm_w + idx);
        float4 v;
        v.x = xv.x * nw.x; v.y = xv.y * nw.y; v.z = xv.z * nw.z; v.w = xv.w * nw.w;
        *(float4*)(s_x + idx) = v;
        acc[0] += xv.x * xv.x + xv.y * xv.y + xv.z * xv.z + xv.w * xv.w;
#pragma unroll
        for (int j = 0; j < 4; ++j) {
            const float4 w = *(const float4*)(w_pre + (size_t)j * DD + idx);
            acc[1 + j] += v.x * w.x + v.y * w.y + v.z * w.z + v.w * w.w;
        }
#pragma unroll
        for (int j = 0; j < 4; ++j) {
            const float4 w = *(const float4*)(w_post + (size_t)j * DD + idx);
            acc[5 + j] += v.x * w.x + v.y * w.y + v.z * w.z + v.w * w.w;
        }
#pragma unroll
        for (int j = 0; j < 16; ++j) {
            const float4 w = *(const float4*)(w_res + (size_t)j * DD + idx);
            acc[9 + j] += v.x * w.x + v.y * w.y + v.z * w.z + v.w * w.w;
        }
    }

    const int lane = tid & 31;
    const int wv   = tid >> 5;
#pragma unroll
    for (int i = 0; i < 25; ++i) {
        float v = acc[i];
#pragma unroll
        for (int o = 16; o > 0; o >>= 1) v += __shfl_down(v, o, 32);
        if (lane == 0) s_red[i][wv] = v;
    }
    __syncthreads();
    if (tid < 25) {
        float v = 0.0f;
#pragma unroll
        for (int w = 0; w < 8; ++w) v += s_red[tid][w];
        s_red[tid][0] = v;
    }
    __syncthreads();
    if (tid == 0) {
        const float rms = sqrtf(s_red[0][0] * (1.0f / (float)DD));
        s_scal[0] = 1.0f / (rms + EPSR);
    }
    __syncthreads();
    const float inv = s_scal[0];
    if (tid < 4) {
        s_scal[1 + tid] = tanhf(a_pre[0] * s_red[1 + tid][0] * inv + b_pre[tid]);
    } else if (tid < 8) {
        const int n = tid - 4;
        h_post_o[(size_t)t * 4 + n] =
            tanhf(a_post[0] * s_red[5 + n][0] * inv + b_post[n]);
    } else if (tid < 24) {
        const int m = tid - 8;
        h_res_o[(size_t)t * 16 + m] =
            tanhf(a_res[0] * s_red[9 + m][0] * inv + b_res[m]);
    }
    __syncthreads();

    const float hp0 = s_scal[1], hp1 = s_scal[2], hp2 = s_scal[3], hp3 = s_scal[4];
    const int c = tid * 4;
    const float4 v0 = *(const float4*)(s_x + c);
    const float4 v1 = *(const float4*)(s_x + CH + c);
    const float4 v2 = *(const float4*)(s_x + 2 * CH + c);
    const float4 v3 = *(const float4*)(s_x + 3 * CH + c);
    float4 r;
    r.x = inv * (hp0 * v0.x + hp1 * v1.x + hp2 * v2.x + hp3 * v3.x);
    r.y = inv * (hp0 * v0.y + hp1 * v1.y + hp2 * v2.y + hp3 * v3.y);
    r.z = inv * (hp0 * v0.z + hp1 * v1.z + hp2 * v2.z + hp3 * v3.z);
    r.w = inv * (hp0 * v0.w + hp1 * v1.w + hp2 * v2.w + hp3 * v3.w);
    *(float4*)(h_in + (size_t)t * CH + c) = r;
}

// ---------------------------------------------------------------------------
// Kernel 2: h_out = h_in @ w_sub^T via V_WMMA_F32_16X16X4_F32, double-buffered
// async global->LDS staging, fused h_res stream-mix + h_post epilogue.
// Block = 64 tokens x 128 channels, 8 waves (2x4), each wave a 32x32 tile.
// ---------------------------------------------------------------------------
__global__ __launch_bounds__(256) void hc_gemm(
    const float* __restrict__ h_in, const float* __restrict__ w_sub,
    const float* __restrict__ x,
    const float* __restrict__ h_post_t, const float* __restrict__ h_res_t,
    float* __restrict__ out)
{
    extern __shared__ float smem[];
    float* sA   = smem + SM_A;    // [2][64 * STRD]
    float* sB   = smem + SM_B;    // [2][128 * STRD]
    float* s_hp = smem + SM_HP;   // 64 tokens x 4
    float* s_hr = smem + SM_HR;   // 64 tokens x 16

    const int tid = threadIdx.x;
    const int t0  = blockIdx.x * 64;
    const int c0  = blockIdx.y * 128;
    const int lane = tid & 31;
    const int wv   = tid >> 5;
    const int tw   = wv >> 2;          // token half of wave (0..1) -> 32 tokens
    const int cwq  = wv & 3;           // channel quarter (0..3)    -> 32 channels
    const int nn   = lane & 15;
    const int hi   = lane >> 4;

    // stage per-token coefficients (64 tokens)
    s_hp[tid] = h_post_t[(size_t)t0 * 4 + tid];
#pragma unroll
    for (int i = 0; i < 4; ++i)
        s_hr[tid + i * 256] = h_res_t[(size_t)t0 * 16 + tid + i * 256];

    // async issue of one K-chunk into buffer bb
    auto issue_chunk = [&](int bb, int k0) {
        // A tile: 64 x KT floats = 1024 16B chunks, 4 per thread
#pragma unroll
        for (int rep = 0; rep < 4; ++rep) {
            const int ci  = tid + rep * 256;
            const int row = ci >> 4, col = (ci & 15) * 4;
            const unsigned lds = (unsigned)(size_t)&sA[bb * A_BUF + row * STRD + col];
            const unsigned go  = (unsigned)(((size_t)(t0 + row) * CH + k0 + col) * 4);
            async_b128(lds, h_in, go);
        }
        // B tile: 128 x KT floats = 2048 16B chunks, 8 per thread (w_sub L2-resident)
#pragma unroll
        for (int rep = 0; rep < 8; ++rep) {
            const int ci  = tid + rep * 256;
            const int row = ci >> 4, col = (ci & 15) * 4;
            const unsigned lds = (unsigned)(size_t)&sB[bb * B_BUF + row * STRD + col];
            const unsigned go  = (unsigned)(((size_t)(c0 + row) * CH + k0 + col) * 4);
            async_b128(lds, w_sub, go);
        }
    };

    v8f acc00 = {}, acc01 = {}, acc10 = {}, acc11 = {};

    issue_chunk(0, 0);
    int cur = 0;
    const int rA0 = (tw * 32 + nn) * STRD;     // A frag base row offset
    const int rB0 = (cwq * 32 + nn) * STRD;    // B frag base row offset

    for (int k0 = 0; k0 < CH; k0 += KT) {
        wait_async0();          // chunk `cur` landed in LDS
        __syncthreads();        // visible to all waves; all waves done with buffer cur^1
        if (k0 + KT < CH) issue_chunk(cur ^ 1, k0 + KT);

        const float* Ab = sA + cur * A_BUF;
        const float* Bb = sB + cur * B_BUF;
        // A 16x4 f32 layout (ISA 7.12.2): lane holds row M=lane%16; VGPR j is K = j + 2*(lane>=16)
#pragma unroll
        for (int kk = 0; kk < KT; kk += 4) {
            const int ca = kk + 2 * hi;
            v2f a0, a1, b0, b1;
            a0[0] = Ab[rA0 + ca];             a0[1] = Ab[rA0 + ca + 1];
            a1[0] = Ab[rA0 + 16 * STRD + ca]; a1[1] = Ab[rA0 + 16 * STRD + ca + 1];
            b0[0] = Bb[rB0 + ca];             b0[1] = Bb[rB0 + ca + 1];
            b1[0] = Bb[rB0 + 16 * STRD + ca]; b1[1] = Bb[rB0 + 16 * STRD + ca + 1];
            acc00 = __builtin_amdgcn_wmma_f32_16x16x4_f32(false, a0, false, b0, (short)0, acc00, false, false);
            acc01 = __builtin_amdgcn_wmma_f32_16x16x4_f32(false, a0, false, b1, (short)0, acc01, false, false);
            acc10 = __builtin_amdgcn_wmma_f32_16x16x4_f32(false, a1, false, b0, (short)0, acc10, false, false);
            acc11 = __builtin_amdgcn_wmma_f32_16x16x4_f32(false, a1, false, b1, (short)0, acc11, false, false);
        }
        cur ^= 1;
    }

    // Epilogue: C/D layout -> VGPR r, lane L: M = r + 8*(L>=16), N = L%16
    v8f accs[2][2] = {{acc00, acc01}, {acc10, acc11}};
    const int cwb = c0 + cwq * 32;
#pragma unroll
    for (int p = 0; p < 2; ++p) {
#pragma unroll
        for (int r = 0; r < 8; ++r) {
            const int lt = tw * 32 + p * 16 + r + 8 * hi;   // local token 0..63
            const int t  = t0 + lt;
#pragma unroll
            for (int q = 0; q < 2; ++q) {
                const int cc = cwb + q * 16 + nn;
                const float h = accs[p][q][r];
                const size_t xb = (size_t)t * NSTR * CH + cc;
                const float x0 = x[xb];
                const float x1 = x[xb + CH];
                const float x2 = x[xb + 2 * CH];
                const float x3 = x[xb + 3 * CH];
#pragma unroll
                for (int n = 0; n < 4; ++n) {
                    out[xb + (size_t)n * CH] =
                          s_hp[lt * 4 + n] * h
                        + s_hr[lt * 16 + n * 4 + 0] * x0
                        + s_hr[lt * 16 + n * 4 + 1] * x1
                        + s_hr[lt * 16 + n * 4 + 2] * x2
                        + s_hr[lt * 16 + n * 4 + 3] * x3;
                }
            }
        }
    }
}

// ---------------------------------------------------------------------------
extern "C" void kernel_launch(void* const* d_in, const int* in_sizes, int n_in,
                              void* d_out, int out_size, void* d_ws, size_t ws_size,
                              hipStream_t stream) {
    const float* x      = (const float*)d_in[0];
    const float* norm_w = (const float*)d_in[1];
    const float* w_pre  = (const float*)d_in[2];
    const float* w_post = (const float*)d_in[3];
    const float* w_res  = (const float*)d_in[4];
    const float* a_pre  = (const float*)d_in[5];
    const float* b_pre  = (const float*)d_in[6];
    const float* a_post = (const float*)d_in[7];
    const float* b_post = (const float*)d_in[8];
    const float* a_res  = (const float*)d_in[9];
    const float* b_res  = (const float*)d_in[10];
    const float* w_sub  = (const float*)d_in[11];
    float* out = (float*)d_out;

    // workspace: h_in [TOK*CH] | h_post [TOK*4] | h_res [TOK*16]  (~34 MB)
    float* h_in     = (float*)d_ws;
    float* h_post_t = h_in + (size_t)TOK * CH;
    float* h_res_t  = h_post_t + (size_t)TOK * 4;

    hc_prep<<<TOK, 256, 0, stream>>>(x, norm_w, w_pre, w_post, w_res,
                                     a_pre, b_pre, a_post, b_post, a_res, b_res,
                                     h_in, h_post_t, h_res_t);

    dim3 grid(TOK / 64, CH / 128);
    hc_gemm<<<grid, 256, SMEM_BYTES, stream>>>(h_in, w_sub, x, h_post_t, h_res_t, out);
}